// Radon3D_33002528702573
// MI455X (gfx1250) — compile-verified
//
#include <hip/hip_runtime.h>

// ---------------------------------------------------------------------------
// Radon3D loss for MI455X (gfx1250).
//   vol: [1,1,64,64,64] f32.  Slice s: im[d,w] = vol[d*4096 + s*64 + w].
//   L = ceil(sqrt(64^2+64^2)) = 91, top = left = 13, angles = linspace(0,120,120) deg.
//   ix = 45*(cos*u - sin*v) + 32 ; iy = 45*(sin*u + cos*v) + 32 ;
//     u = (j-45)/45 (output column), v = (i-45)/45 (canvas row, summed over).
//   Incremental per row step (dv = 1/45): d_ix = -sin, d_iy = +cos.
// ---------------------------------------------------------------------------

#define RD_D   64
#define RD_W   64
#define RD_S   64
#define RD_A   120
#define RD_L   91
#define RD_ACH 8                       // angles per block
#define RD_NCH (RD_A / RD_ACH)         // 15
#define RD_NBLK (RD_S * RD_NCH)        // 960
#define RD_PITCH 68                    // floats; 68*4 = 272 bytes, 16B aligned rows

typedef __attribute__((ext_vector_type(2))) float v2f;
typedef __attribute__((ext_vector_type(8))) float v8f;

typedef int v4i __attribute__((vector_size(16)));
typedef __attribute__((address_space(1))) v4i Gv4i;   // global AS v4i
typedef __attribute__((address_space(3))) v4i Lv4i;   // LDS AS v4i
typedef __attribute__((address_space(3))) char LasC;

static __device__ __forceinline__ void async_copy_b128(const float* g, float* l) {
#if __has_builtin(__builtin_amdgcn_global_load_async_to_lds_b128)
  __builtin_amdgcn_global_load_async_to_lds_b128((Gv4i*)g, (Lv4i*)l, 0, 0);
#else
  unsigned loff = (unsigned)(__SIZE_TYPE__)(LasC*)l;
  asm volatile("global_load_async_to_lds_b128 %0, %1, off"
               :: "v"(loff), "v"(g) : "memory");
#endif
}

static __device__ __forceinline__ void wait_async0() {
#if __has_builtin(__builtin_amdgcn_s_wait_asynccnt)
  __builtin_amdgcn_s_wait_asynccnt(0);
#else
  asm volatile("s_wait_asynccnt 0" ::: "memory");
#endif
}

__global__ __launch_bounds__(256)
void radon3d_partial(const float* __restrict__ outv,
                     const float* __restrict__ gtv,
                     float* __restrict__ partial) {
  __shared__ __align__(16) float sO[RD_D * RD_PITCH];
  __shared__ __align__(16) float sG[RD_D * RD_PITCH];
  __shared__ float red[256];

  const int tid = threadIdx.x;
  const int blk = blockIdx.x;
  const int s   = blk / RD_NCH;        // slice index
  const int ch  = blk % RD_NCH;        // angle chunk

  // ---- Stage slice s of both volumes into LDS via async global->LDS DMA ----
  // One slice = 64 rows x 256 bytes = 1024 b128 transfers; 256 lanes x 4 iters.
  #pragma unroll
  for (int k = 0; k < 4; ++k) {
    int t = tid + (k << 8);
    int d = t >> 4;                    // row 0..63
    int q = t & 15;                    // 16B quad within row
    const int goff = d * (RD_W * RD_S) + s * RD_W + q * 4;
    const int loff = d * RD_PITCH + q * 4;
    async_copy_b128(outv + goff, &sO[loff]);
    async_copy_b128(gtv  + goff, &sG[loff]);
  }
  wait_async0();
  __syncthreads();

  // ---- Line integrals: each thread handles (a, x) pairs ----
  const float DTH = (float)(3.14159265358979323846 * (120.0 / 119.0) / 180.0);
  float lsum = 0.0f;

  for (int it = tid; it < RD_ACH * RD_L; it += 256) {
    const int al = it / RD_L;
    const int j  = it - al * RD_L;     // output column
    const int a  = ch * RD_ACH + al;

    const float theta = (float)a * DTH;
    float sn, cs;
    __sincosf(theta, &sn, &cs);

    const float u = (float)(j - 45) * (1.0f / 45.0f);
    // i = 0 (v = -1):
    float ix = fmaf(45.0f * cs, u,  45.0f * sn + 32.0f);
    float iy = fmaf(45.0f * sn, u, -45.0f * cs + 32.0f);

    float accO = 0.0f, accG = 0.0f;
    for (int i = 0; i < RD_L; ++i) {
      const float fy = floorf(iy), fx = floorf(ix);
      const int y0 = (int)fy, x0 = (int)fx;
      const float dy = iy - fy, dx = ix - fx;

      // weights masked to zero outside [0,63]; addresses clamped.
      const float wy0 = ((unsigned)y0       < 64u) ? (1.0f - dy) : 0.0f;
      const float wy1 = ((unsigned)(y0 + 1) < 64u) ? dy          : 0.0f;
      const float wx0 = ((unsigned)x0       < 64u) ? (1.0f - dx) : 0.0f;
      const float wx1 = ((unsigned)(x0 + 1) < 64u) ? dx          : 0.0f;

      const int y0c = max(0, min(63, y0));
      const int y1c = max(0, min(63, y0 + 1));
      const int x0c = max(0, min(63, x0));
      const int x1c = max(0, min(63, x0 + 1));

      const int r0 = y0c * RD_PITCH, r1 = y1c * RD_PITCH;
      const float o = wy0 * (wx0 * sO[r0 + x0c] + wx1 * sO[r0 + x1c]) +
                      wy1 * (wx0 * sO[r1 + x0c] + wx1 * sO[r1 + x1c]);
      const float g = wy0 * (wx0 * sG[r0 + x0c] + wx1 * sG[r0 + x1c]) +
                      wy1 * (wx0 * sG[r1 + x0c] + wx1 * sG[r1 + x1c]);
      accO += o;
      accG += g;

      ix -= sn;                        // v += 1/45  =>  ix += -45*sn/45
      iy += cs;                        //               iy +=  45*cs/45
    }
    lsum += fabsf(accO - accG);
  }

  red[tid] = lsum;
  __syncthreads();

#if __has_builtin(__builtin_amdgcn_wmma_f32_16x16x4_f32)
  // ---- Block reduction on the matrix pipe: D = A(16x4) * ones(4x16) + C,
  // accumulated over 4 chunks of 64 values; C[m,n] ends up as chunk row-sums.
  if ((tid >> 5) == 0) {               // wave 0 only; wave-uniform => EXEC all 1s
    const int lane = tid & 31;
    const int m  = lane & 15;
    const int k0 = (lane >> 4) << 1;   // lanes 0-15: K=0,1 ; lanes 16-31: K=2,3
    v2f b; b[0] = 1.0f; b[1] = 1.0f;   // all-ones B (layout-independent)
    v8f c = {};
    #pragma unroll
    for (int cb = 0; cb < 4; ++cb) {
      v2f aa;
      aa[0] = red[cb * 64 + m * 4 + k0];
      aa[1] = red[cb * 64 + m * 4 + k0 + 1];
      c = __builtin_amdgcn_wmma_f32_16x16x4_f32(false, aa, false, b,
                                                (short)0, c, false, false);
    }
    // Lane holds C[M, N=lane%16] for M = v (+8 if lane>=16); all columns equal.
    float h = c[0] + c[1] + c[2] + c[3] + c[4] + c[5] + c[6] + c[7];
    h += __shfl_xor(h, 16, 32);        // add the other 8 rows
    if (lane == 0) partial[blk] = h;
  }
#else
  for (int off = 128; off > 0; off >>= 1) {
    if (tid < off) red[tid] += red[tid + off];
    __syncthreads();
  }
  if (tid == 0) partial[blk] = red[0];
#endif
}

__global__ __launch_bounds__(256)
void radon3d_final(const float* __restrict__ partial, float* __restrict__ out) {
  __shared__ float red[256];
  const int tid = threadIdx.x;
  float s = 0.0f;
  for (int i = tid; i < RD_NBLK; i += 256) s += partial[i];
  red[tid] = s;
  __syncthreads();
  for (int off = 128; off > 0; off >>= 1) {
    if (tid < off) red[tid] += red[tid + off];
    __syncthreads();
  }
  if (tid == 0) out[0] = red[0] * (1.0f / (float)(RD_A * RD_L));
}

extern "C" void kernel_launch(void* const* d_in, const int* in_sizes, int n_in,
                              void* d_out, int out_size, void* d_ws, size_t ws_size,
                              hipStream_t stream) {
  (void)in_sizes; (void)n_in; (void)out_size; (void)ws_size;
  const float* outv = (const float*)d_in[0];
  const float* gtv  = (const float*)d_in[1];
  float* partial = (float*)d_ws;                 // RD_NBLK floats of scratch
  float* loss    = (float*)d_out;                // single f32

  radon3d_partial<<<RD_NBLK, 256, 0, stream>>>(outv, gtv, partial);
  radon3d_final<<<1, 256, 0, stream>>>(partial, loss);
}